// NWkernelRegiresion_35003983463189
// MI455X (gfx1250) — compile-verified
//
#include <hip/hip_runtime.h>

typedef float f4 __attribute__((ext_vector_type(4)));
typedef int v4i __attribute__((ext_vector_type(4)));

#define BLOCK 256                    // 8 waves of 32 on gfx1250
#define CHUNK 2048                   // floats per matrix per pipeline stage (8 KB)
#define VEC_PER_STAGE (CHUNK / 4)    // 512 float4 per matrix per stage
#define IOPT (VEC_PER_STAGE / BLOCK) // 2 b128 issues per thread per matrix

#if defined(__AMDGCN__) && __has_builtin(__builtin_amdgcn_global_load_async_to_lds_b128)
#define HAVE_ASYNC 1
#else
#define HAVE_ASYNC 0
#endif

#if HAVE_ASYNC
__device__ __forceinline__ void async_cp_b128(const f4* g, const float* l) {
  // (v4i AS1* global src, v4i AS3* lds dst, imm offset, imm cpol)
  __builtin_amdgcn_global_load_async_to_lds_b128(
      (__attribute__((address_space(1))) v4i*)g,
      (__attribute__((address_space(3))) v4i*)l,
      0, 0);
}

__device__ __forceinline__ void wait_async_le4() {
#if __has_builtin(__builtin_amdgcn_s_wait_asynccnt)
  __builtin_amdgcn_s_wait_asynccnt(4);
#else
  asm volatile("s_wait_asynccnt 0x4" ::: "memory");
#endif
}
__device__ __forceinline__ void wait_async_0() {
#if __has_builtin(__builtin_amdgcn_s_wait_asynccnt)
  __builtin_amdgcn_s_wait_asynccnt(0);
#else
  asm volatile("s_wait_asynccnt 0x0" ::: "memory");
#endif
}
#endif // HAVE_ASYNC

// ---- online softmax primitives -------------------------------------------

__device__ __forceinline__ void online_merge(float& m, float& s, float& a,
                                             float m2, float s2, float a2) {
  float nm = fmaxf(m, m2);
  float c1 = __expf(m - nm);
  float c2 = __expf(m2 - nm);
  s = fmaf(s, c1, s2 * c2);
  a = fmaf(a, c1, a2 * c2);
  m = nm;
}

__device__ __forceinline__ void accum1(float q, float w, float kv, float vv,
                                       float& m, float& s, float& a) {
  float d = (q - kv) * w;
  float l = -0.5f * d * d;
  float nm = fmaxf(m, l);
  float c = __expf(m - nm);   // m == -inf only before first element -> exp(-inf)=0
  float e = __expf(l - nm);
  s = fmaf(s, c, e);
  a = fmaf(a, c, e * vv);
  m = nm;
}

__device__ __forceinline__ void accum4(float q, float w, f4 kk, f4 vv,
                                       float& m, float& s, float& a) {
#pragma unroll
  for (int j = 0; j < 4; ++j) accum1(q, w, kk[j], vv[j], m, s, a);
}

// wave32 shuffle reduce + cross-wave LDS reduce; result valid on thread 0
__device__ __forceinline__ float block_reduce_out(float m, float s, float a) {
#pragma unroll
  for (int off = 16; off > 0; off >>= 1) {
    float m2 = __shfl_xor(m, off, 32);
    float s2 = __shfl_xor(s, off, 32);
    float a2 = __shfl_xor(a, off, 32);
    online_merge(m, s, a, m2, s2, a2);
  }
  __shared__ float rm[BLOCK / 32], rs[BLOCK / 32], ra[BLOCK / 32];
  const int lane = threadIdx.x & 31;
  const int wid = threadIdx.x >> 5;
  if (lane == 0) { rm[wid] = m; rs[wid] = s; ra[wid] = a; }
  __syncthreads();
  float r = 0.0f;
  if (threadIdx.x < (BLOCK / 32)) {
    m = rm[threadIdx.x]; s = rs[threadIdx.x]; a = ra[threadIdx.x];
#pragma unroll
    for (int off = (BLOCK / 32) / 2; off > 0; off >>= 1) {
      float m2 = __shfl_xor(m, off, BLOCK / 32);
      float s2 = __shfl_xor(s, off, BLOCK / 32);
      float a2 = __shfl_xor(a, off, BLOCK / 32);
      online_merge(m, s, a, m2, s2, a2);
    }
    r = a / s;
  }
  return r;
}

// direct-from-HBM streaming body (fallback / non-multiple K)
__device__ __forceinline__ void stream_plain(const float* kr, const float* vr,
                                             int K, float q, float w,
                                             float& m, float& s, float& a) {
  const f4* k4 = (const f4*)kr;
  const f4* v4 = (const f4*)vr;
  const int nv = K >> 2;
  for (int i = threadIdx.x; i < nv; i += BLOCK) {
    f4 kk = __builtin_nontemporal_load(&k4[i]);  // NT: 512MB read-once, keep out of L2
    f4 vv = __builtin_nontemporal_load(&v4[i]);
    accum4(q, w, kk, vv, m, s, a);
  }
  for (int i = (nv << 2) + threadIdx.x; i < K; i += BLOCK)
    accum1(q, w, kr[i], vr[i], m, s, a);
}

// ---- kernels (async kernel first so the disasm snippet shows it) ----------

// async-to-LDS double-buffered pipeline; requires K % CHUNK == 0
__global__ __launch_bounds__(BLOCK) void nw_async_kernel(
    const float* __restrict__ q, const float* __restrict__ keys,
    const float* __restrict__ vals, const float* __restrict__ wp,
    float* __restrict__ out, int K) {
  const int n = blockIdx.x;
  const float qv = q[n];
  const float w = wp[0];
  float m = -__builtin_inff(), s = 0.f, a = 0.f;
  const float* kr = keys + (size_t)n * K;
  const float* vr = vals + (size_t)n * K;
#if HAVE_ASYNC
  __shared__ __align__(16) float lk[2][CHUNK];
  __shared__ __align__(16) float lv[2][CHUNK];
  const int tid = threadIdx.x;
  const int stages = K / CHUNK;

  // stage issue: 2 b128 per thread per matrix => 4 async ops per wave per stage
  auto issue = [&](int t, int buf) {
    const f4* gk = (const f4*)kr + t * VEC_PER_STAGE;
    const f4* gv = (const f4*)vr + t * VEC_PER_STAGE;
#pragma unroll
    for (int j = 0; j < IOPT; ++j) {
      int i = tid + j * BLOCK;
      async_cp_b128(&gk[i], &lk[buf][i * 4]);
      async_cp_b128(&gv[i], &lv[buf][i * 4]);
    }
  };

  issue(0, 0);
  for (int t = 0; t < stages; ++t) {
    const int buf = t & 1;
    if (t + 1 < stages) {
      issue(t + 1, buf ^ 1);  // prefetch next stage into other buffer
      wait_async_le4();       // in-order completion => stage t has landed
    } else {
      wait_async_0();
    }
    __syncthreads();          // publish all waves' async LDS writes
    const f4* sk = (const f4*)lk[buf];
    const f4* sv = (const f4*)lv[buf];
#pragma unroll
    for (int j = 0; j < IOPT; ++j) {
      int i = tid + j * BLOCK;
      accum4(qv, w, sk[i], sv[i], m, s, a);
    }
    __syncthreads();          // done reading buf before it is refilled
  }
#else
  stream_plain(kr, vr, K, qv, w, m, s, a);
#endif
  float r = block_reduce_out(m, s, a);
  if (threadIdx.x == 0) out[n] = r;
}

__global__ __launch_bounds__(BLOCK) void nw_plain_kernel(
    const float* __restrict__ q, const float* __restrict__ keys,
    const float* __restrict__ vals, const float* __restrict__ wp,
    float* __restrict__ out, int K) {
  const int n = blockIdx.x;
  const float qv = q[n];
  const float w = wp[0];
  float m = -__builtin_inff(), s = 0.f, a = 0.f;
  stream_plain(keys + (size_t)n * K, vals + (size_t)n * K, K, qv, w, m, s, a);
  float r = block_reduce_out(m, s, a);
  if (threadIdx.x == 0) out[n] = r;
}

// ---- host launch ----------------------------------------------------------

extern "C" void kernel_launch(void* const* d_in, const int* in_sizes, int n_in,
                              void* d_out, int out_size, void* d_ws, size_t ws_size,
                              hipStream_t stream) {
  const float* q    = (const float*)d_in[0];
  const float* keys = (const float*)d_in[1];
  const float* vals = (const float*)d_in[2];
  const float* w    = (const float*)d_in[3];
  float* out = (float*)d_out;
  const int N = in_sizes[0];
  const int K = in_sizes[1] / N;   // 8192 x 8192
  if (N <= 0) return;
  if ((K % CHUNK) == 0) {
    nw_async_kernel<<<N, BLOCK, 0, stream>>>(q, keys, vals, w, out, K);
  } else {
    nw_plain_kernel<<<N, BLOCK, 0, stream>>>(q, keys, vals, w, out, K);
  }
}